// RoutingRNN_10402410791508
// MI455X (gfx1250) — compile-verified
//
#include <hip/hip_runtime.h>

#define B_      16
#define N_INP   256
#define N_OUT   64
#define D_COV   8
#define D_INP   16
#define D_OUT   16
#define HID     64
#define CH      (D_COV * D_OUT)   /* 128 */
#define EPSV    1e-5f

typedef float v2f __attribute__((ext_vector_type(2)));
typedef float v8f __attribute__((ext_vector_type(8)));

__device__ __forceinline__ float sigmoidf_(float x) { return 1.0f / (1.0f + expf(-x)); }
__device__ __forceinline__ float logsigmoidf_(float x) {
    // stable log(sigmoid(x)) = min(x,0) - log1p(exp(-|x|))
    return fminf(x, 0.0f) - log1pf(expf(-fabsf(x)));
}

// -----------------------------------------------------------------------------
// Kernel 1: V[b,i,j,c,h] = sum_d W[i,j,d,h] * mu[b,i,c,d]   (bias added at store)
// Per i: GEMM [128 x 16] @ [16 x 1024] using V_WMMA_F32_16X16X4_F32 (full f32).
// Wave tile: 16 rows x 64 cols (4 j's), 4 K-steps -> 16 WMMA per wave.
// -----------------------------------------------------------------------------
__global__ void __launch_bounds__(256) k_votes_wmma(
    const float* __restrict__ mu,    // [B, N_INP, D_COV, D_INP]
    const float* __restrict__ W,     // [N_INP, N_OUT, D_INP, D_OUT]
    const float* __restrict__ bias,  // [N_INP, N_OUT, D_COV, D_OUT]
    float* __restrict__ V)           // [B, N_INP, N_OUT, D_COV, D_OUT]
{
    const int i    = blockIdx.x;                 // 0..255
    const int wave = threadIdx.x >> 5;           // 0..7
    const int lane = threadIdx.x & 31;
    const int wt   = blockIdx.y * 8 + wave;      // 0..127 wave tiles
    const int mTile = wt & 7;                    // 8 tiles of 16 rows (b,c)
    const int nTile = wt >> 3;                   // 16 tiles of 64 cols (4 j)
    const int j0   = nTile * 4;
    const int lo   = lane & 15;
    const int hi   = lane >> 4;

    // A fragment addressing: row = mTile*16 + lo maps to (b = row>>3, c = row&7)
    const int rowA = mTile * 16 + lo;
    const int bA   = rowA >> 3;
    const int cA   = rowA & 7;
    const float* Abase = mu + (size_t)bA * (N_INP * D_COV * D_INP)
                            + (size_t)i  * (D_COV * D_INP)
                            + (size_t)cA * D_INP;
    const float* Wbase = W + (size_t)i * (N_OUT * D_INP * D_OUT);

    v8f acc[4];
    acc[0] = (v8f){}; acc[1] = (v8f){}; acc[2] = (v8f){}; acc[3] = (v8f){};

#pragma unroll
    for (int kk = 0; kk < 4; ++kk) {
        const int k = kk * 4 + 2 * hi;           // lanes 0-15: K=k0,k0+1; 16-31: K=k0+2,k0+3
        v2f a;
        a.x = Abase[k];
        a.y = Abase[k + 1];
        const float* wb = Wbase + k * D_OUT + lo;   // W[i, j, k, h=lo]
#pragma unroll
        for (int jj = 0; jj < 4; ++jj) {
            const float* wj = wb + (size_t)(j0 + jj) * (D_INP * D_OUT);
            v2f bf;
            bf.x = wj[0];
            bf.y = wj[D_OUT];                    // K=k+1 row
            acc[jj] = __builtin_amdgcn_wmma_f32_16x16x4_f32(
                false, a, false, bf, (short)0, acc[jj], false, false);
        }
    }

    // Store + bias.  C/D layout: lane, vgpr v -> (M = v + 8*hi, N = lo).
    // row = mTile*16 + v + 8*hi  ->  b = row>>3,  c = row&7 == v.
    const float* bb = bias + (size_t)i * (N_OUT * CH);
#pragma unroll
    for (int v = 0; v < 8; ++v) {
        const int row  = mTile * 16 + v + 8 * hi;
        const int bOut = row >> 3;
        const int cOut = v;
        const size_t vbase = (size_t)bOut * ((size_t)N_INP * N_OUT * CH)
                           + (size_t)i * (N_OUT * CH)
                           + (size_t)cOut * D_OUT + lo;
#pragma unroll
        for (int jj = 0; jj < 4; ++jj) {
            const int j = j0 + jj;
            V[vbase + (size_t)j * CH] = acc[jj][v] + bb[(size_t)j * CH + cOut * D_OUT + lo];
        }
    }
}

// -----------------------------------------------------------------------------
// Iteration 0 routing: D[b,i,j] = sigmoid(a_inp[b,i]) / N_OUT
// -----------------------------------------------------------------------------
__global__ void k_D0(const float* __restrict__ a_inp, float* __restrict__ D)
{
    int idx = blockIdx.x * blockDim.x + threadIdx.x;
    if (idx >= B_ * N_INP * N_OUT) return;
    int bi = idx / N_OUT;                    // b*N_INP + i
    D[idx] = sigmoidf_(a_inp[bi]) * (1.0f / N_OUT);
}

// -----------------------------------------------------------------------------
// Iterations >=1 routing: softmax over j of logsig(a_out) - logdet - q,
// q = sum_ch (V - mu)^2 * (0.5/sig2).   One block per (i,b), 64 threads (j).
// -----------------------------------------------------------------------------
__global__ void __launch_bounds__(64) k_R(
    const float* __restrict__ V,
    const float* __restrict__ a_inp,
    const float* __restrict__ a_out,    // [B, N_OUT]
    const float* __restrict__ mu_o,     // [B, N_OUT, CH]
    const float* __restrict__ hinv,     // [B, N_OUT, CH]  = 0.5/sig2
    const float* __restrict__ logdet,   // [B, N_OUT]      = 0.5*sum log sig2
    float* __restrict__ D)
{
    __shared__ float sRed[N_OUT];
    const int i = blockIdx.x;
    const int b = blockIdx.y;
    const int j = threadIdx.x;

    const float* vb = V + (size_t)b * ((size_t)N_INP * N_OUT * CH)
                        + (size_t)i * (N_OUT * CH)
                        + (size_t)j * CH;
    const float* mb = mu_o + (size_t)(b * N_OUT + j) * CH;
    const float* hb = hinv + (size_t)(b * N_OUT + j) * CH;

    float q = 0.0f;
#pragma unroll 4
    for (int t = 0; t < CH; ++t) {
        float dv = vb[t] - mb[t];
        q += dv * dv * hb[t];
    }
    float s = logsigmoidf_(a_out[b * N_OUT + j]) - logdet[b * N_OUT + j] - q;

    // softmax over the 64 j's
    sRed[j] = s;
    __syncthreads();
    for (int off = 32; off >= 1; off >>= 1) {
        if (j < off) sRed[j] = fmaxf(sRed[j], sRed[j + off]);
        __syncthreads();
    }
    float mx = sRed[0];
    __syncthreads();
    float e = expf(s - mx);
    sRed[j] = e;
    __syncthreads();
    for (int off = 32; off >= 1; off >>= 1) {
        if (j < off) sRed[j] += sRed[j + off];
        __syncthreads();
    }
    float sum = sRed[0];
    float fa = sigmoidf_(a_inp[b * N_INP + i]);
    D[(size_t)(b * N_INP + i) * N_OUT + j] = fa * e / sum;
}

// -----------------------------------------------------------------------------
// Per-(b,j) reductions over i: Dsum and a_temp (= sum (bu+bi)*D - sum bi*fa)
// -----------------------------------------------------------------------------
__global__ void k_reduce_bj(
    const float* __restrict__ D, const float* __restrict__ a_inp,
    const float* __restrict__ beta_use, const float* __restrict__ beta_ign,
    float* __restrict__ Dsum, float* __restrict__ a_temp)
{
    int idx = blockIdx.x * blockDim.x + threadIdx.x;
    if (idx >= B_ * N_OUT) return;
    int b = idx / N_OUT;
    int j = idx % N_OUT;
    float ds = 0.0f, at = 0.0f;
    for (int i = 0; i < N_INP; ++i) {
        float d  = D[(size_t)(b * N_INP + i) * N_OUT + j];
        float fa = sigmoidf_(a_inp[b * N_INP + i]);
        float u  = beta_use[i * N_OUT + j];
        float g  = beta_ign[i * N_OUT + j];
        ds += d;
        at += (u + g) * d - g * fa;
    }
    Dsum[idx]   = ds;
    a_temp[idx] = at;
}

// -----------------------------------------------------------------------------
// LSTM cell + output head, one block per batch element.
// -----------------------------------------------------------------------------
__global__ void __launch_bounds__(256) k_lstm(
    const float* __restrict__ a_temp,
    const float* __restrict__ w_ih, const float* __restrict__ w_hh,
    const float* __restrict__ b_ih, const float* __restrict__ b_hh,
    const float* __restrict__ w_out, const float* __restrict__ b_out,
    float* __restrict__ h, float* __restrict__ c,
    float* __restrict__ a_out, float* __restrict__ out_a)
{
    __shared__ float sA[HID], sH[HID], sG[4 * HID];
    const int b = blockIdx.x;
    const int t = threadIdx.x;
    if (t < HID) { sA[t] = a_temp[b * N_OUT + t]; sH[t] = h[b * HID + t]; }
    __syncthreads();
    {
        float acc = b_ih[t] + b_hh[t];
        const float* wi = w_ih + (size_t)t * N_OUT;
        const float* wh = w_hh + (size_t)t * HID;
#pragma unroll 4
        for (int k = 0; k < HID; ++k) acc += sA[k] * wi[k] + sH[k] * wh[k];
        sG[t] = acc;
    }
    __syncthreads();
    if (t < HID) {
        float gi = sG[t], gf = sG[HID + t], gg = sG[2 * HID + t], go = sG[3 * HID + t];
        float cn = sigmoidf_(gf) * c[b * HID + t] + sigmoidf_(gi) * tanhf(gg);
        float hn = sigmoidf_(go) * tanhf(cn);
        c[b * HID + t] = cn;
        h[b * HID + t] = hn;
        sH[t] = hn;
    }
    __syncthreads();
    if (t < N_OUT) {
        float acc = b_out[t];
        const float* wo = w_out + (size_t)t * HID;
#pragma unroll 4
        for (int k = 0; k < HID; ++k) acc += sH[k] * wo[k];
        a_out[b * N_OUT + t] = acc;
        out_a[b * N_OUT + t] = acc;   // final iteration's write is the output
    }
}

// -----------------------------------------------------------------------------
// Moments: one block per (j,b), 128 threads over (c,h).  Single V pass:
//   mu   = (sum_i D*V) * over
//   sig2 = (sum_i D*V^2)*over - mu^2*(2 - Dsum*over) + EPS   (exact rewrite)
// Also emits 0.5/sig2 and logdet for the next iteration's routing pass.
// -----------------------------------------------------------------------------
__global__ void __launch_bounds__(128) k_moments(
    const float* __restrict__ V, const float* __restrict__ D,
    const float* __restrict__ Dsum,
    float* __restrict__ mu_o, float* __restrict__ sig_o,
    float* __restrict__ hinv, float* __restrict__ logdet,
    float* __restrict__ out_mu, float* __restrict__ out_sig, int last)
{
    __shared__ float sD[N_INP];
    __shared__ float sRed[128];
    const int j = blockIdx.x;
    const int b = blockIdx.y;
    const int t = threadIdx.x;

    sD[t]       = D[(size_t)(b * N_INP + t) * N_OUT + j];
    sD[t + 128] = D[(size_t)(b * N_INP + t + 128) * N_OUT + j];
    __syncthreads();

    const float* vb = V + (size_t)b * ((size_t)N_INP * N_OUT * CH)
                        + (size_t)j * CH + t;
    float acc = 0.0f, acc2 = 0.0f;
#pragma unroll 4
    for (int i = 0; i < N_INP; ++i) {
        float d = sD[i];
        float v = vb[(size_t)i * (N_OUT * CH)];
        acc  += d * v;
        acc2 += d * v * v;
    }
    float dsum = Dsum[b * N_OUT + j];
    float over = 1.0f / (dsum + EPSV);
    float m    = acc * over;
    float S    = dsum * over;
    float s2   = acc2 * over - m * m * (2.0f - S) + EPSV;

    const int o = (b * N_OUT + j) * CH + t;
    mu_o[o]  = m;
    sig_o[o] = s2;
    hinv[o]  = 0.5f / s2;
    if (last) { out_mu[o] = m; out_sig[o] = s2; }

    sRed[t] = logf(s2);
    __syncthreads();
    for (int off = 64; off >= 1; off >>= 1) {
        if (t < off) sRed[t] += sRed[t + off];
        __syncthreads();
    }
    if (t == 0) logdet[b * N_OUT + j] = 0.5f * sRed[0];
}

__global__ void k_init(float* __restrict__ h, float* __restrict__ c)
{
    int t = blockIdx.x * blockDim.x + threadIdx.x;
    if (t < B_ * HID) { h[t] = 0.0f; c[t] = 0.0f; }
}

// -----------------------------------------------------------------------------
extern "C" void kernel_launch(void* const* d_in, const int* in_sizes, int n_in,
                              void* d_out, int out_size, void* d_ws, size_t ws_size,
                              hipStream_t stream)
{
    const float* a_inp    = (const float*)d_in[0];
    const float* mu_inp   = (const float*)d_in[1];
    const float* W        = (const float*)d_in[2];
    const float* B_bias   = (const float*)d_in[3];
    const float* beta_use = (const float*)d_in[4];
    const float* beta_ign = (const float*)d_in[5];
    const float* w_ih     = (const float*)d_in[6];
    const float* w_hh     = (const float*)d_in[7];
    const float* b_ih     = (const float*)d_in[8];
    const float* b_hh     = (const float*)d_in[9];
    const float* w_out    = (const float*)d_in[10];
    const float* b_out    = (const float*)d_in[11];

    float* ws = (float*)d_ws;
    size_t o = 0;
    float* V      = ws + o; o += (size_t)B_ * N_INP * N_OUT * CH;  // 33.5M floats
    float* D      = ws + o; o += (size_t)B_ * N_INP * N_OUT;
    float* Dsum   = ws + o; o += B_ * N_OUT;
    float* a_temp = ws + o; o += B_ * N_OUT;
    float* a_out  = ws + o; o += B_ * N_OUT;
    float* h      = ws + o; o += B_ * HID;
    float* c      = ws + o; o += B_ * HID;
    float* mu_o   = ws + o; o += (size_t)B_ * N_OUT * CH;
    float* sig_o  = ws + o; o += (size_t)B_ * N_OUT * CH;
    float* hinv   = ws + o; o += (size_t)B_ * N_OUT * CH;
    float* logdet = ws + o; o += B_ * N_OUT;

    float* out_a   = (float*)d_out;
    float* out_mu  = out_a + B_ * N_OUT;
    float* out_sig = out_mu + (size_t)B_ * N_OUT * CH;

    k_init<<<dim3((B_ * HID + 255) / 256), 256, 0, stream>>>(h, c);

    // V = einsum + bias, WMMA f32 GEMM (256 i-slices x 16 blocks of 8 waves)
    k_votes_wmma<<<dim3(N_INP, 16), 256, 0, stream>>>(mu_inp, W, B_bias, V);

    for (int it = 0; it < 3; ++it) {
        if (it == 0)
            k_D0<<<dim3((B_ * N_INP * N_OUT + 255) / 256), 256, 0, stream>>>(a_inp, D);
        else
            k_R<<<dim3(N_INP, B_), 64, 0, stream>>>(V, a_inp, a_out, mu_o, hinv, logdet, D);

        k_reduce_bj<<<dim3((B_ * N_OUT + 255) / 256), 256, 0, stream>>>(
            D, a_inp, beta_use, beta_ign, Dsum, a_temp);

        k_lstm<<<dim3(B_), 256, 0, stream>>>(
            a_temp, w_ih, w_hh, b_ih, b_hh, w_out, b_out, h, c, a_out, out_a);

        k_moments<<<dim3(N_OUT, B_), 128, 0, stream>>>(
            V, D, Dsum, mu_o, sig_o, hinv, logdet, out_mu, out_sig, (it == 2) ? 1 : 0);
    }
}